// TorchLSTM_78580721647903
// MI455X (gfx1250) — compile-verified
//
#include <hip/hip_runtime.h>
#include <math.h>

// Problem constants (match reference).
#define BB   64
#define TT   512
#define II   256
#define HH   256
#define G4   1024   // 4*H

typedef __attribute__((ext_vector_type(2))) float v2f;
typedef __attribute__((ext_vector_type(4))) float v4f;
typedef __attribute__((ext_vector_type(8))) float v8f;

__device__ __forceinline__ v8f wmma_f32(v2f a, v2f b, v8f c) {
    return __builtin_amdgcn_wmma_f32_16x16x4_f32(false, a, false, b, (short)0, c,
                                                 false, false);
}

__device__ __forceinline__ float sigmoidf_(float x) {
    return 1.0f / (1.0f + __expf(-x));
}

// ---------------------------------------------------------------------------
// Kernel 0: zero-init h ping-pong buffers and c state.
// ---------------------------------------------------------------------------
__global__ void zero_kernel(float* __restrict__ p, int n) {
    int i = blockIdx.x * blockDim.x + threadIdx.x;
    if (i < n) p[i] = 0.0f;
}

// ---------------------------------------------------------------------------
// Kernel 1: x_proj[BT, 4H] = x[BT, I] @ W_ih^T + (b_ih + b_hh)
// Block: 256 threads (8 waves), tile 64(M) x 64(N), K chunked by 32 via LDS.
// K-major LDS layouts: b128 staging stores, b64 operand-pair loads.
// ---------------------------------------------------------------------------
__global__ __launch_bounds__(256) void xproj_kernel(
    const float* __restrict__ x, const float* __restrict__ W_ih,
    const float* __restrict__ b_ih, const float* __restrict__ b_hh,
    float* __restrict__ xproj) {
    __shared__ float As[64][36];   // [m][k]  pitch 36: b64 reads conflict-free
    __shared__ float Bs[64][36];   // [n][k]  K-major so (k0,k0+1) contiguous

    const int tid    = threadIdx.x;
    const int blockM = blockIdx.x * 64;
    const int blockN = blockIdx.y * 64;
    const int w      = tid >> 5;
    const int lane   = tid & 31;
    const int m      = lane & 15;        // A row / B col within tile
    const int hi     = lane >> 4;        // half-wave: K offset +2, M offset +8
    const int rowb   = (w & 3) * 16;     // wave M offset within block tile
    const int colb   = (w >> 2) * 32;    // wave N offset within block tile

    v8f acc0 = {}; v8f acc1 = {};

    for (int kc = 0; kc < II; kc += 32) {
        // Stage A chunk 64x32 (b128 global load -> b128 LDS store).
        int e = tid * 4;
        #pragma unroll
        for (int p = 0; p < 2; ++p) {
            int row = e >> 5, col = e & 31;
            v4f v = *(const v4f*)(x + (size_t)(blockM + row) * II + kc + col);
            *(v4f*)&As[row][col] = v;
            e += 1024;
        }
        // Stage B chunk: W_ih is [4H, I] row-major; keep K-major in LDS.
        e = tid * 4;
        #pragma unroll
        for (int p = 0; p < 2; ++p) {
            int row = e >> 5, col = e & 31;   // row = n (0..63), col = k
            v4f v = *(const v4f*)(W_ih + (size_t)(blockN + row) * II + kc + col);
            *(v4f*)&Bs[row][col] = v;
            e += 1024;
        }
        __syncthreads();

        #pragma unroll
        for (int kk = 0; kk < 32; kk += 4) {
            const int k0 = kk + hi * 2;
            v2f a  = *(const v2f*)&As[rowb + m][k0];
            v2f b0 = *(const v2f*)&Bs[colb + m][k0];
            acc0 = wmma_f32(a, b0, acc0);
            v2f b1 = *(const v2f*)&Bs[colb + 16 + m][k0];
            acc1 = wmma_f32(a, b1, acc1);
        }
        __syncthreads();
    }

    const int gc0 = blockN + colb + m;
    const int gc1 = gc0 + 16;
    const float bias0 = b_ih[gc0] + b_hh[gc0];
    const float bias1 = b_ih[gc1] + b_hh[gc1];
    #pragma unroll
    for (int r = 0; r < 8; ++r) {
        const int gm = blockM + rowb + r + hi * 8;
        xproj[(size_t)gm * G4 + gc0] = acc0[r] + bias0;
        xproj[(size_t)gm * G4 + gc1] = acc1[r] + bias1;
    }
}

// ---------------------------------------------------------------------------
// Kernel 2: one LSTM timestep.
// Grid: (B/16, H/64) = 4x4 blocks, 256 threads each.
// Block computes gates[16 batches, 4 gates x 64 cols] = xproj_t + h @ W_hh^T,
// then fused activations / state update. h ping-pongs across launches.
// ---------------------------------------------------------------------------
__global__ __launch_bounds__(256) void lstm_step_kernel(
    const float* __restrict__ xproj, const float* __restrict__ W_hh,
    const float* __restrict__ h_in, float* __restrict__ h_out,
    float* __restrict__ c_st, float* __restrict__ y,
    float* __restrict__ hT, float* __restrict__ cT, int t, int last) {
    __shared__ float Hs[16][260];    // h tile [batch][k], full K=256 (pitch 260)
    __shared__ float Ws[256][36];    // W_hh^T chunk [gateconcat][k], K-major

    const int tid    = threadIdx.x;
    const int blockB = blockIdx.x * 16;
    const int blockN = blockIdx.y * 64;
    const int w      = tid >> 5;
    const int lane   = tid & 31;
    const int m      = lane & 15;
    const int hi     = lane >> 4;
    const int q      = w >> 1;                    // gate handled by this wave
    const int colb   = q * 64 + (w & 1) * 32;     // col in 256-wide gate concat

    // Stage full h tile (16 x 256) once; b128 stores.
    {
        int e = tid * 4;
        #pragma unroll
        for (int p = 0; p < 4; ++p) {
            int row = e >> 8, col = e & 255;
            v4f v = *(const v4f*)(h_in + (size_t)(blockB + row) * HH + col);
            *(v4f*)&Hs[row][col] = v;
            e += 1024;
        }
    }

    v8f acc0 = {}; v8f acc1 = {};
    for (int kc = 0; kc < HH; kc += 32) {
        // Stage W_hh^T chunk: 256 gate-concat cols x 32 K, K-major.
        int e = tid * 4;
        #pragma unroll
        for (int p = 0; p < 8; ++p) {
            int cidx = e >> 5;                    // 0..255 gate-concat col
            int k    = e & 31;
            int qq = cidx >> 6, nl = cidx & 63;
            int gc = qq * HH + blockN + nl;       // row of W_hh [4H, H]
            v4f v = *(const v4f*)(W_hh + (size_t)gc * HH + kc + k);
            *(v4f*)&Ws[cidx][k] = v;
            e += 1024;
        }
        __syncthreads();   // covers Hs stage on first iteration too

        #pragma unroll
        for (int kk = 0; kk < 32; kk += 4) {
            const int kl = kk + hi * 2;
            const int kg = kc + kl;
            v2f a  = *(const v2f*)&Hs[m][kg];
            v2f b0 = *(const v2f*)&Ws[colb + m][kl];
            acc0 = wmma_f32(a, b0, acc0);
            v2f b1 = *(const v2f*)&Ws[colb + 16 + m][kl];
            acc1 = wmma_f32(a, b1, acc1);
        }
        __syncthreads();
    }

    // Exchange gate tile through LDS (reuse Ws storage, stride 257).
    float* Gs = &Ws[0][0];
    #pragma unroll
    for (int r = 0; r < 8; ++r) {
        Gs[(r + hi * 8) * 257 + colb + m]      = acc0[r];
        Gs[(r + hi * 8) * 257 + colb + 16 + m] = acc1[r];
    }
    __syncthreads();

    // Fused activations + state update. 16x64 elements, 4 per thread.
    #pragma unroll
    for (int p = 0; p < 4; ++p) {
        const int e  = tid + p * 256;
        const int mm = e >> 6, nl = e & 63;
        const int gb = blockB + mm;
        const int col = blockN + nl;
        const size_t xrow = ((size_t)gb * TT + t) * G4;
        float gi = Gs[mm * 257 + nl]       + xproj[xrow + 0 * HH + col];
        float gf = Gs[mm * 257 + 64 + nl]  + xproj[xrow + 1 * HH + col];
        float gg = Gs[mm * 257 + 128 + nl] + xproj[xrow + 2 * HH + col];
        float go = Gs[mm * 257 + 192 + nl] + xproj[xrow + 3 * HH + col];
        const float iv = sigmoidf_(gi);
        const float fv = sigmoidf_(gf);
        const float gv = tanhf(gg);
        const float ov = sigmoidf_(go);
        const size_t ci = (size_t)gb * HH + col;
        const float cn = fv * c_st[ci] + iv * gv;
        const float hn = ov * tanhf(cn);
        c_st[ci]  = cn;
        h_out[ci] = hn;
        y[((size_t)gb * TT + t) * HH + col] = hn;
        if (last) { hT[ci] = hn; cT[ci] = cn; }
    }
}

// ---------------------------------------------------------------------------
extern "C" void kernel_launch(void* const* d_in, const int* in_sizes, int n_in,
                              void* d_out, int out_size, void* d_ws, size_t ws_size,
                              hipStream_t stream) {
    (void)in_sizes; (void)n_in; (void)out_size; (void)ws_size;
    const float* x    = (const float*)d_in[0];
    const float* W_ih = (const float*)d_in[1];
    const float* W_hh = (const float*)d_in[2];
    const float* b_ih = (const float*)d_in[3];
    const float* b_hh = (const float*)d_in[4];

    float* out = (float*)d_out;
    float* y   = out;                              // [B, T, H]
    float* hT  = y + (size_t)BB * TT * HH;         // [1, B, H]
    float* cT  = hT + (size_t)BB * HH;             // [1, B, H]

    float* ws    = (float*)d_ws;
    float* xproj = ws;                             // [B*T, 4H] = 128 MB
    float* h0    = xproj + (size_t)BB * TT * G4;   // [B, H]
    float* h1    = h0 + (size_t)BB * HH;           // [B, H]
    float* cst   = h1 + (size_t)BB * HH;           // [B, H]

    // Zero h ping-pong + c state (d_ws is poisoned, and timed replays reuse it).
    {
        const int n = 3 * BB * HH;
        zero_kernel<<<(n + 255) / 256, 256, 0, stream>>>(h0, n);
    }

    // Phase 1: input projection GEMM (parallel, off the critical path).
    {
        dim3 grid((BB * TT) / 64, G4 / 64);        // 512 x 16
        xproj_kernel<<<grid, 256, 0, stream>>>(x, W_ih, b_ih, b_hh, xproj);
    }

    // Phase 2: sequential scan; stream ordering provides the step dependency.
    for (int t = 0; t < TT; ++t) {
        const float* hin  = (t & 1) ? h1 : h0;
        float*       hout = (t & 1) ? h0 : h1;
        dim3 grid(BB / 16, HH / 64);               // 4 x 4
        lstm_step_kernel<<<grid, 256, 0, stream>>>(xproj, W_hh, hin, hout, cst,
                                                   y, hT, cT, t, t == TT - 1);
    }
}